// CrossModalPositionalEncoding_48902497632813
// MI455X (gfx1250) — compile-verified
//
#include <hip/hip_runtime.h>
#include <hip/hip_bf16.h>

// ---------------------------------------------------------------------------
// CrossModalPositionalEncoding for MI455X (gfx1250, wave32)
//
// out[b,t,:] = tok[b,t,:] + pos_emb[mid[b,t], local[b,t], :]
// local[b,t] = #earlier tokens in row b with the same modality (4 modalities).
//
// Memory-bound (~0.12 FLOP/byte, ~280 MB HBM traffic -> ~12 us at 23.3 TB/s):
//  - Phase 1: per-row 4-way segmented prefix count via LDS int4 scan (tiny).
//  - Phase 2: 4 tokens per block to amortize wave-launch overhead; b128
//    streams with non-temporal hints so the 16 MB pos_emb table stays
//    resident in the 192 MB L2; global_prefetch_b8 issued for all 4 gathered
//    rows before consumption to hide L2 latency.
// ---------------------------------------------------------------------------

typedef float f4 __attribute__((ext_vector_type(4)));

#define T_LEN       2048
#define MAX_SEQ     1024
#define D_MODEL     1024
#define BDIM        256
#define CHUNK       (T_LEN / BDIM)      // 8 tokens per thread
#define F4_PER_TOK  (D_MODEL / 4)       // 256 -> one float4 per thread
#define TOK_PER_BLK 4

// ---------------- Phase 1: per-row modality-local position scan -------------
__global__ __launch_bounds__(BDIM)
void cmpe_scan_kernel(const int* __restrict__ mids, int* __restrict__ rowidx) {
    const int b   = blockIdx.x;
    const int tid = threadIdx.x;
    const int* __restrict__ row = mids + b * T_LEN;
    const int base = tid * CHUNK;

    // Per-thread counts of each modality within its chunk.
    int m[CHUNK];
    int c0 = 0, c1 = 0, c2 = 0, c3 = 0;
#pragma unroll
    for (int i = 0; i < CHUNK; ++i) {
        const int mm = row[base + i];
        m[i] = mm;
        c0 += (mm == 0); c1 += (mm == 1); c2 += (mm == 2); c3 += (mm == 3);
    }

    // Hillis-Steele inclusive scan of int4 counts across the 256 threads.
    __shared__ int4 sc[2][BDIM];
    int in = 0;
    sc[0][tid] = make_int4(c0, c1, c2, c3);
    __syncthreads();
#pragma unroll
    for (int off = 1; off < BDIM; off <<= 1) {
        int4 v = sc[in][tid];
        if (tid >= off) {
            const int4 u = sc[in][tid - off];
            v.x += u.x; v.y += u.y; v.z += u.z; v.w += u.w;
        }
        sc[in ^ 1][tid] = v;
        in ^= 1;
        __syncthreads();
    }
    // Exclusive prefix for this thread's chunk.
    int o0 = 0, o1 = 0, o2 = 0, o3 = 0;
    if (tid > 0) {
        const int4 e = sc[in][tid - 1];
        o0 = e.x; o1 = e.y; o2 = e.z; o3 = e.w;
    }

    // Replay chunk: emit packed gather row index mid*MAX_SEQ + local.
#pragma unroll
    for (int i = 0; i < CHUNK; ++i) {
        const int mm = m[i];
        int local = (mm == 0) ? o0 : (mm == 1) ? o1 : (mm == 2) ? o2 : o3;
        o0 += (mm == 0); o1 += (mm == 1); o2 += (mm == 2); o3 += (mm == 3);
        if (local > MAX_SEQ - 1) local = MAX_SEQ - 1;   // JAX clamp semantics
        rowidx[b * T_LEN + base + i] = mm * MAX_SEQ + local;
    }
}

// ---------------- Phase 2: gather + add, 4 tokens per block -----------------
__global__ __launch_bounds__(BDIM)
void cmpe_add_kernel(const f4* __restrict__ tok,
                     const f4* __restrict__ pos,
                     const int* __restrict__ rowidx,
                     f4* __restrict__ out) {
    const int    tid      = threadIdx.x;
    const size_t base_tok = (size_t)blockIdx.x * TOK_PER_BLK;

    // 4 uniform packed row indices -> single scalar s_load_b128.
    const int4 r4 = *(const int4*)(rowidx + base_tok);
    const int  rs0 = r4.x, rs1 = r4.y, rs2 = r4.z, rs3 = r4.w;

    // Prefetch all 4 gathered pos_emb rows up front (global_prefetch_b8),
    // so rows 1..3 are in flight while row 0 is consumed.
    if (tid == 0) __builtin_prefetch(pos + (size_t)rs0 * F4_PER_TOK, 0, 0);
    if (tid == 1) __builtin_prefetch(pos + (size_t)rs1 * F4_PER_TOK, 0, 0);
    if (tid == 2) __builtin_prefetch(pos + (size_t)rs2 * F4_PER_TOK, 0, 0);
    if (tid == 3) __builtin_prefetch(pos + (size_t)rs3 * F4_PER_TOK, 0, 0);

    const int rr[TOK_PER_BLK] = {rs0, rs1, rs2, rs3};
#pragma unroll
    for (int k = 0; k < TOK_PER_BLK; ++k) {
        const f4* __restrict__ pr = pos + (size_t)rr[k] * F4_PER_TOK;
        const size_t off = (base_tok + k) * F4_PER_TOK + tid;

        // Streaming operands: non-temporal so pos_emb stays hot in L2.
        f4 a = __builtin_nontemporal_load(&tok[off]);
        a += pr[tid];                         // RT policy: keep table in L2
        __builtin_nontemporal_store(a, &out[off]);
    }
}

// ---------------------------------------------------------------------------
extern "C" void kernel_launch(void* const* d_in, const int* in_sizes, int n_in,
                              void* d_out, int out_size, void* d_ws, size_t ws_size,
                              hipStream_t stream) {
    const float* tok  = (const float*)d_in[0];   // (B, T, D) f32
    const int*   mids = (const int*)d_in[1];     // (B, T)    int
    const float* pos  = (const float*)d_in[2];   // (M, 1024, D) f32
    float*       out  = (float*)d_out;           // (B, T, D) f32
    int*         rowidx = (int*)d_ws;            // (B*T) packed gather indices

    const int BT = in_sizes[1];                  // B*T = 32768
    const int Bn = BT / T_LEN;                   // B  = 16

    cmpe_scan_kernel<<<dim3(Bn), dim3(BDIM), 0, stream>>>(mids, rowidx);
    cmpe_add_kernel<<<dim3(BT / TOK_PER_BLK), dim3(BDIM), 0, stream>>>(
        (const f4*)tok, (const f4*)pos, rowidx, (f4*)out);
}